// SO3net_55637006353155
// MI455X (gfx1250) — compile-verified
//
#include <hip/hip_runtime.h>
#include <math.h>

#define MTOT 9
#define NF 32
#define NRBF 20

// ============================================================
// Compile-time real Clebsch-Gordan table (matches reference's
// _build_cg_real for LMAX=2). All constexpr so device code sees
// only immediates after template unrolling.
// ============================================================
constexpr double cfact(int n){ double r=1.0; for(int i=2;i<=n;++i) r*=(double)i; return r; }
constexpr double csqrt_(double x){ if(x<=0.0) return 0.0; double g = x>1.0?x:1.0; for(int i=0;i<32;++i) g=0.5*(g+x/g); return g; }
constexpr double S2C = 0.70710678118654752440; // 1/sqrt(2)

constexpr double cg_c(int j1,int m1,int j2,int m2,int j3,int m3){
  if(m1+m2!=m3) return 0.0;
  int lo = j1>j2 ? j1-j2 : j2-j1;
  if(j3<lo || j3>j1+j2) return 0.0;
  double pre = (2.0*j3+1.0)*cfact(j1+j2-j3)*cfact(j1-j2+j3)*cfact(-j1+j2+j3)/cfact(j1+j2+j3+1);
  pre = csqrt_(pre) * csqrt_(cfact(j3+m3)*cfact(j3-m3)*cfact(j1-m1)*cfact(j1+m1)*cfact(j2-m2)*cfact(j2+m2));
  double s=0.0;
  for(int k=0;k<=j1+j2+j3;++k){
    int d1=j1+j2-j3-k, d2=j1-m1-k, d3=j2+m2-k, d4=j3-j2+m1+k, d5=j3-j1-m2+k;
    if(d1<0||d2<0||d3<0||d4<0||d5<0) continue;
    double t = cfact(k)*cfact(d1)*cfact(d2)*cfact(d3)*cfact(d4)*cfact(d5);
    s += ((k&1)? -1.0 : 1.0)/t;
  }
  return pre*s;
}

struct Cplx { double re; double im; };
constexpr Cplx ureal(int l,int a,int i){
  Cplx u{0.0,0.0};
  if(a==l){ if(i==l) u.re=1.0; return u; }
  if(a>l){ int m=a-l;
    if(i==l+m)      u.re = (m&1)? -S2C : S2C;
    else if(i==l-m) u.re = S2C;
  } else { int m=l-a;
    if(i==l-m)      u.im = S2C;
    else if(i==l+m) u.im = (m&1)? S2C : -S2C; // -i*(-1)^m/sqrt2
  }
  return u;
}

constexpr int l_of(int idx){ return idx==0?0:(idx<4?1:2); }
constexpr int a_of(int idx){ int l=l_of(idx); return idx - l*l; }

constexpr double cg_real(int i1,int i2,int i3){
  int l1=l_of(i1), l2=l_of(i2), l3=l_of(i3);
  int a=a_of(i1), b=a_of(i2), c=a_of(i3);
  double re=0.0;
  for(int i=0;i<=2*l1;++i){ Cplx u1=ureal(l1,a,i); if(u1.re==0.0&&u1.im==0.0) continue;
    for(int j=0;j<=2*l2;++j){ Cplx u2=ureal(l2,b,j); if(u2.re==0.0&&u2.im==0.0) continue;
      for(int k=0;k<=2*l3;++k){ Cplx u3=ureal(l3,c,k); if(u3.re==0.0&&u3.im==0.0) continue;
        double C = cg_c(l1,i-l1,l2,j-l2,l3,k-l3); if(C==0.0) continue;
        double ar=u1.re*u2.re-u1.im*u2.im, ai=u1.re*u2.im+u1.im*u2.re;
        re += (ar*u3.re + ai*u3.im)*C;   // (u1*u2*conj(u3)).re * C
      }}}
  return re;
}

constexpr int MAXT=512;
struct CGT{ int n; int i1[MAXT]; int i2[MAXT]; int io[MAXT]; int fl[MAXT]; float v[MAXT]; };
constexpr CGT build_cg(){
  CGT t{};
  for(int a=0;a<MTOT;++a)
    for(int b=0;b<MTOT;++b)
      for(int o=0;o<MTOT;++o){
        double val = cg_real(a,b,o);
        if(val>1e-10 || val<-1e-10){
          if(t.n<MAXT){ t.i1[t.n]=a; t.i2[t.n]=b; t.io[t.n]=o; t.fl[t.n]=l_of(b); t.v[t.n]=(float)val; ++t.n; }
        }
      }
  return t;
}
constexpr CGT CG = build_cg();
constexpr int NCG = CG.n;

// Unrolled CG contractions (indices/values fold to immediates).
template<int T>
__device__ inline void cg_msg(float acc[MTOT], const float xj[MTOT], const float Yv[MTOT], const float Wv[3]){
  if constexpr (T < NCG){
    constexpr int a=CG.i1[T], b=CG.i2[T], o=CG.io[T], l=CG.fl[T];
    constexpr float v=CG.v[T];
    acc[o] = fmaf(xj[a]*Yv[b], v*Wv[l], acc[o]);
    cg_msg<T+1>(acc, xj, Yv, Wv);
  }
}
template<int T>
__device__ inline void cg_selfi(float acc[MTOT], const float a9[MTOT], const float b9[MTOT]){
  if constexpr (T < NCG){
    constexpr int a=CG.i1[T], b=CG.i2[T], o=CG.io[T];
    constexpr float v=CG.v[T];
    acc[o] = fmaf(a9[a]*b9[b], v, acc[o]);
    cg_selfi<T+1>(acc, a9, b9);
  }
}

// ============================================================
// WMMA f32 GEMM: D = A[MxK] @ B[KxN] (+epilogue), 16x16 tile/wave
// ============================================================
typedef __attribute__((ext_vector_type(2))) float v2f;
typedef __attribute__((ext_vector_type(8))) float v8f;

// mode 0: out = AB   1: out = (AB + bias[col])*scale[row]
// mode 2: out = AB + bias[col]   3: out += AB
__global__ void wmma_gemm_f32(const float* __restrict__ A, const float* __restrict__ B,
                              float* __restrict__ out,
                              const float* __restrict__ bias, const float* __restrict__ scale,
                              int M, int N, int K, int lda, int ldb, int ldo, int mode)
{
  int wave = blockIdx.x * (blockDim.x >> 5) + (threadIdx.x >> 5);
  int mtiles = (M + 15) >> 4;
  int ntiles = N >> 4;
  if (wave >= mtiles * ntiles) return;        // wave-uniform exit (EXEC stays all-ones)
  int mt = wave / ntiles, nt = wave % ntiles;
  int lane = threadIdx.x & 31;
  int half = lane >> 4;
  int lr   = lane & 15;
  int arow = mt*16 + lr;  if (arow >= M) arow = M-1;   // clamp loads, keep EXEC full
  int bcol = nt*16 + lr;

  v8f c = {};
  for (int k = 0; k < K; k += 4){
    v2f a, b;
    a.x = A[(size_t)arow*lda + k + 2*half    ];
    a.y = A[(size_t)arow*lda + k + 2*half + 1];
    b.x = B[(size_t)(k + 2*half    )*ldb + bcol];
    b.y = B[(size_t)(k + 2*half + 1)*ldb + bcol];
    c = __builtin_amdgcn_wmma_f32_16x16x4_f32(false, a, false, b, (short)0, c, false, false);
  }
  #pragma unroll
  for (int v = 0; v < 8; ++v){
    int r = mt*16 + half*8 + v;
    if (r < M){
      float val = c[v];
      if (mode == 1)      val = (val + bias[bcol]) * scale[r];
      else if (mode == 2) val =  val + bias[bcol];
      if (mode == 3) out[(size_t)r*ldo + bcol] += val;
      else           out[(size_t)r*ldo + bcol]  = val;
    }
  }
}

// ============================================================
// Per-edge geometry: d, dir, Y(9), RBF(20), cutoff
// ============================================================
__global__ void geom_kernel(const float* __restrict__ r, float* __restrict__ Y,
                            float* __restrict__ RB, float* __restrict__ CUT, int E)
{
  int e = blockIdx.x*blockDim.x + threadIdx.x;
  if (e >= E) return;
  float rx = r[3*e], ry = r[3*e+1], rz = r[3*e+2];
  float d  = sqrtf(rx*rx + ry*ry + rz*rz);
  float inv = 1.0f/d;
  float x = rx*inv, y = ry*inv, z = rz*inv;
  const float c0=0.28209479177387814f, c1=0.4886025119029199f;
  const float c2=1.0925484305920792f, c20=0.31539156525252005f, c22=0.5462742152960396f;
  float* Yp = Y + (size_t)e*MTOT;
  Yp[0]=c0; Yp[1]=c1*y; Yp[2]=c1*z; Yp[3]=c1*x;
  Yp[4]=c2*x*y; Yp[5]=c2*y*z; Yp[6]=c20*(3.0f*z*z-1.0f); Yp[7]=c2*x*z; Yp[8]=c22*(x*x-y*y);
  const float spacing = 5.0f/(NRBF-1);
  const float coeff = -0.5f/(spacing*spacing);
  float* Rp = RB + (size_t)e*NRBF;
  #pragma unroll
  for (int k = 0; k < NRBF; ++k){ float t = d - spacing*(float)k; Rp[k] = expf(coeff*t*t); }
  CUT[e] = (d < 5.0f) ? 0.5f*(cosf(d*0.6283185307179586f) + 1.0f) : 0.0f;
}

// ============================================================
// x init: x[n,0,:] = emb[Z[n]], rest 0       (x lives in d_out)
// ============================================================
__global__ void init_x_kernel(const int* __restrict__ Z, const float* __restrict__ emb,
                              float* __restrict__ x, int total)
{
  int t = blockIdx.x*blockDim.x + threadIdx.x;
  if (t >= total) return;
  int f = t & 31; int m = (t>>5) % MTOT; int n = t / (MTOT*NF);
  x[t] = (m == 0) ? emb[(size_t)Z[n]*NF + f] : 0.0f;
}

__global__ void zero_kernel(float* __restrict__ p, int n){
  int t = blockIdx.x*blockDim.x + threadIdx.x;
  if (t < n) p[t] = 0.0f;
}

// ============================================================
// Edge message: one wave32 per edge, lane = feature channel.
// acc[m] = sum_t CG contraction; atomic scatter to dx[idx_i].
// ============================================================
__global__ void message_kernel(const float* __restrict__ x, const float* __restrict__ Y,
                               const float* __restrict__ W, const int* __restrict__ idx_i,
                               const int* __restrict__ idx_j, float* __restrict__ dx, int E)
{
  int e = blockIdx.x*(blockDim.x>>5) + (threadIdx.x>>5);
  if (e >= E) return;
  int f = threadIdx.x & 31;
  int i = idx_i[e], j = idx_j[e];
  const float* Yp = Y + (size_t)e*MTOT;
  float Yv[MTOT];
  #pragma unroll
  for (int m = 0; m < MTOT; ++m) Yv[m] = Yp[m];
  float Wv[3];
  #pragma unroll
  for (int l = 0; l < 3; ++l) Wv[l] = W[(size_t)e*96 + l*NF + f];
  float xj[MTOT];
  #pragma unroll
  for (int m = 0; m < MTOT; ++m) xj[m] = x[(size_t)j*(MTOT*NF) + m*NF + f];
  float acc[MTOT];
  #pragma unroll
  for (int m = 0; m < MTOT; ++m) acc[m] = 0.0f;
  cg_msg<0>(acc, xj, Yv, Wv);
  #pragma unroll
  for (int m = 0; m < MTOT; ++m)
    atomicAdd(&dx[(size_t)i*(MTOT*NF) + m*NF + f], acc[m]);
}

// ============================================================
// Node self-interaction: dxp = dx + CG(dx, ddx)
// ============================================================
__global__ void selfcg_kernel(const float* __restrict__ dx, const float* __restrict__ ddx,
                              float* __restrict__ dxp, int Nn)
{
  int n = blockIdx.x*(blockDim.x>>5) + (threadIdx.x>>5);
  if (n >= Nn) return;
  int f = threadIdx.x & 31;
  float a9[MTOT], b9[MTOT], acc[MTOT];
  #pragma unroll
  for (int m = 0; m < MTOT; ++m){
    a9[m] = dx [(size_t)n*(MTOT*NF) + m*NF + f];
    b9[m] = ddx[(size_t)n*(MTOT*NF) + m*NF + f];
    acc[m] = 0.0f;
  }
  cg_selfi<0>(acc, a9, b9);
  #pragma unroll
  for (int m = 0; m < MTOT; ++m)
    dxp[(size_t)n*(MTOT*NF) + m*NF + f] = a9[m] + acc[m];
}

// ============================================================
// Gate: dx2[n,m,f] *= sigmoid(h[n, l(m), f])
// ============================================================
__global__ void gate_kernel(const float* __restrict__ h, float* __restrict__ dx2, int total)
{
  int t = blockIdx.x*blockDim.x + threadIdx.x;
  if (t >= total) return;
  int f = t & 31; int m = (t>>5) % MTOT; int n = t / (MTOT*NF);
  int l = (m == 0) ? 0 : (m < 4 ? 1 : 2);
  float hv = h[(size_t)n*96 + l*NF + f];
  dx2[t] *= 1.0f/(1.0f + expf(-hv));
}

// ============================================================
// Host driver
// ============================================================
static inline int ceildiv(int a, int b){ return (a + b - 1)/b; }

static void launch_gemm(const float* A, const float* B, float* out,
                        const float* bias, const float* scale,
                        int M, int N, int K, int lda, int ldb, int ldo, int mode,
                        hipStream_t stream)
{
  int mtiles = (M + 15)/16, ntiles = N/16;
  int waves = mtiles*ntiles;
  hipLaunchKernelGGL(wmma_gemm_f32, dim3(ceildiv(waves, 8)), dim3(256), 0, stream,
                     A, B, out, bias, scale, M, N, K, lda, ldb, ldo, mode);
}

extern "C" void kernel_launch(void* const* d_in, const int* in_sizes, int n_in,
                              void* d_out, int out_size, void* d_ws, size_t ws_size,
                              hipStream_t stream)
{
  const int*   Z      = (const int*)  d_in[0];
  const float* r_ij   = (const float*)d_in[1];
  const int*   idx_i  = (const int*)  d_in[2];
  const int*   idx_j  = (const int*)  d_in[3];
  const float* emb    = (const float*)d_in[4];
  const float* filt_w = (const float*)d_in[5];
  const float* filt_b = (const float*)d_in[6];
  const float* mix1_w = (const float*)d_in[7];
  const float* mix2_w = (const float*)d_in[8];
  const float* mix3_w = (const float*)d_in[9];
  const float* gate_w = (const float*)d_in[10];
  const float* gate_b = (const float*)d_in[11];

  const int Nn = in_sizes[0];
  const int E  = in_sizes[1] / 3;
  const int L  = in_sizes[5] / (NRBF * 3 * NF);

  float* x = (float*)d_out;               // [Nn, 9, 32] accumulated in place

  float* ws = (float*)d_ws;  size_t o = 0;
  float* Y   = ws + o; o += (size_t)E * MTOT;
  float* RB  = ws + o; o += (size_t)E * NRBF;
  float* CUT = ws + o; o += (size_t)E;
  float* W   = ws + o; o += (size_t)E * 96;
  float* dx  = ws + o; o += (size_t)Nn * MTOT * NF;
  float* ddx = ws + o; o += (size_t)Nn * MTOT * NF;
  float* dxp = ws + o; o += (size_t)Nn * MTOT * NF;
  float* dx2 = ws + o; o += (size_t)Nn * MTOT * NF;
  float* h   = ws + o; o += (size_t)Nn * 96;

  const int nodeElems = Nn * MTOT * NF;

  hipLaunchKernelGGL(geom_kernel, dim3(ceildiv(E, 256)), dim3(256), 0, stream, r_ij, Y, RB, CUT, E);
  hipLaunchKernelGGL(init_x_kernel, dim3(ceildiv(nodeElems, 256)), dim3(256), 0, stream, Z, emb, x, nodeElems);

  for (int li = 0; li < L; ++li){
    // W = (RBF @ filt_w + filt_b) * cut      [E, 96]   (WMMA)
    launch_gemm(RB, filt_w + (size_t)li*NRBF*96, W, filt_b + (size_t)li*96, CUT,
                E, 96, NRBF, NRBF, 96, 96, 1, stream);
    // dx = 0; scatter messages
    hipLaunchKernelGGL(zero_kernel, dim3(ceildiv(nodeElems, 256)), dim3(256), 0, stream, dx, nodeElems);
    hipLaunchKernelGGL(message_kernel, dim3(ceildiv(E, 8)), dim3(256), 0, stream,
                       x, Y, W, idx_i, idx_j, dx, E);
    // ddx = dx @ mix1                        [9N, 32]  (WMMA)
    launch_gemm(dx, mix1_w + (size_t)li*NF*NF, ddx, nullptr, nullptr,
                Nn*MTOT, NF, NF, NF, NF, NF, 0, stream);
    // dxp = dx + CG(dx, ddx)
    hipLaunchKernelGGL(selfcg_kernel, dim3(ceildiv(Nn, 8)), dim3(256), 0, stream, dx, ddx, dxp, Nn);
    // dx2 = dxp @ mix2                       (WMMA)
    launch_gemm(dxp, mix2_w + (size_t)li*NF*NF, dx2, nullptr, nullptr,
                Nn*MTOT, NF, NF, NF, NF, NF, 0, stream);
    // h = dx2[:,0,:] @ gate_w + gate_b       [N, 96]   (WMMA, lda=288)
    launch_gemm(dx2, gate_w + (size_t)li*NF*96, h, gate_b + (size_t)li*96, nullptr,
                Nn, 96, NF, MTOT*NF, 96, 96, 2, stream);
    // dx2 *= sigmoid(h[:, l(m), :])
    hipLaunchKernelGGL(gate_kernel, dim3(ceildiv(nodeElems, 256)), dim3(256), 0, stream, h, dx2, nodeElems);
    // x += dx2 @ mix3                        (WMMA, accumulate)
    launch_gemm(dx2, mix3_w + (size_t)li*NF*NF, x, nullptr, nullptr,
                Nn*MTOT, NF, NF, NF, NF, NF, 3, stream);
  }
}